// RouteGNN_25357486916017
// MI455X (gfx1250) — compile-verified
//
#include <hip/hip_runtime.h>
#include <hip/hip_bf16.h>

typedef float v2f __attribute__((ext_vector_type(2)));
typedef float v8f __attribute__((ext_vector_type(8)));

#define NN 100000      // nodes
#define NE 3200000     // edges (before self loops)
#define NG 128         // graphs
#define FIN 16
#define HID 64

static inline unsigned cdiv(long long a, long long b) { return (unsigned)((a + b - 1) / b); }

// ---------------- elementwise helpers ----------------
__global__ void zero_kernel(float* __restrict__ p, long long n) {
  long long i = (long long)blockIdx.x * blockDim.x + threadIdx.x;
  if (i < n) p[i] = 0.0f;
}

__global__ void degree_kernel(const int* __restrict__ dst, float* __restrict__ deg,
                              int nE, int nTot) {
  int e = blockIdx.x * blockDim.x + threadIdx.x;
  if (e >= nTot) return;
  int d = (e < nE) ? dst[e] : (e - nE);
  atomicAdd(&deg[d], 1.0f);
}

__global__ void rsqrt_kernel(float* __restrict__ deg, int n) {
  int i = blockIdx.x * blockDim.x + threadIdx.x;
  if (i < n) deg[i] = rsqrtf(fmaxf(deg[i], 1.0f));
}

__global__ void norm_kernel(const int* __restrict__ src, const int* __restrict__ dst,
                            const float* __restrict__ dinv, float* __restrict__ norm,
                            int nE, int nTot) {
  int e = blockIdx.x * blockDim.x + threadIdx.x;
  if (e >= nTot) return;
  int s, d;
  if (e < nE) { s = src[e]; d = dst[e]; } else { s = e - nE; d = s; }
  norm[e] = dinv[s] * dinv[d];
}

// ---------------- fp32 WMMA GEMM: H[M,64] = X[M,K] @ W[K,64] ----------------
// One wave computes a 16(M) x 64(N) strip via 4 accumulators, K-loop step 4.
// A 16x4 frag: lanes 0-15 row=lane hold K={k0,k0+1}; lanes 16-31 hold K={k0+2,k0+3}.
// B 4x16 frag mirrors A with N in place of M. D 16x16: VGPR j = rows j / j+8.
__global__ void gemm_wmma_kernel(const float* __restrict__ X, const float* __restrict__ W,
                                 float* __restrict__ H, int M, int K) {
  const int lane = threadIdx.x & 31;
  const int wave = threadIdx.x >> 5;
  const int m0 = (blockIdx.x * 4 + wave) * 16;   // wave-uniform M tile
  if (m0 >= M) return;                            // uniform exit: EXEC all-1 in live waves
  const int half = lane >> 4;                     // 0 or 1
  const int l16  = lane & 15;

  v8f acc[4] = {v8f{}, v8f{}, v8f{}, v8f{}};
  for (int k0 = 0; k0 < K; k0 += 4) {
    const float* arow = X + (size_t)(m0 + l16) * K + (k0 + half * 2);
    v2f a; a.x = arow[0]; a.y = arow[1];
    const float* b0 = W + (size_t)(k0 + half * 2) * HID;       // row k0+2*half
    const float* b1 = b0 + HID;                                // row k0+2*half+1
#pragma unroll
    for (int t = 0; t < 4; ++t) {
      v2f b; b.x = b0[l16 + 16 * t]; b.y = b1[l16 + 16 * t];
      acc[t] = __builtin_amdgcn_wmma_f32_16x16x4_f32(
          /*neg_a=*/false, a, /*neg_b=*/false, b,
          /*c_mod=*/(short)0, acc[t], /*reuse_a=*/false, /*reuse_b=*/false);
    }
  }
#pragma unroll
  for (int j = 0; j < 8; ++j) {
    float* orow = H + (size_t)(m0 + j + half * 8) * HID + l16;
    orow[0]  = acc[0][j];
    orow[16] = acc[1][j];
    orow[32] = acc[2][j];
    orow[48] = acc[3][j];
  }
}

// ---------------- edge gather/scatter: AGG[dst] += norm * H[src] ----------------
// 16 threads per edge, float4 per thread (64 feats). Self loops are edges [nE, nTot).
__global__ void edge_scatter_kernel(const float* __restrict__ H,
                                    const int* __restrict__ src, const int* __restrict__ dst,
                                    const float* __restrict__ norm,
                                    float* __restrict__ AGG, int nE, int nTot) {
  long long idx = (long long)blockIdx.x * blockDim.x + threadIdx.x;
  if (idx >= (long long)nTot * 16) return;
  int e = (int)(idx >> 4);
  int c = (int)(idx & 15);
  int s, d;
  if (e < nE) { s = src[e]; d = dst[e]; } else { s = e - nE; d = s; }
  float w = norm[e];
  const float4 v = *reinterpret_cast<const float4*>(H + (size_t)s * HID + c * 4);
  float* out = AGG + (size_t)d * HID + c * 4;
  atomicAdd(out + 0, w * v.x);
  atomicAdd(out + 1, w * v.y);
  atomicAdd(out + 2, w * v.z);
  atomicAdd(out + 3, w * v.w);
}

__global__ void bias_relu_kernel(const float* __restrict__ AGG, const float* __restrict__ b,
                                 float* __restrict__ OUT, long long n) {
  long long i = (long long)blockIdx.x * blockDim.x + threadIdx.x;
  if (i < n) OUT[i] = fmaxf(AGG[i] + b[i & (HID - 1)], 0.0f);
}

// ---------------- global mean pool ----------------
__global__ void pool_kernel(const float* __restrict__ H, const int* __restrict__ batch,
                            float* __restrict__ pooled, float* __restrict__ cnt, int nNodes) {
  long long i = (long long)blockIdx.x * blockDim.x + threadIdx.x;
  if (i >= (long long)nNodes * HID) return;
  int node = (int)(i >> 6);
  int f = (int)(i & (HID - 1));
  int g = batch[node];
  atomicAdd(&pooled[(size_t)g * HID + f], H[i]);
  if (f == 0) atomicAdd(&cnt[g], 1.0f);
}

// ---------------- head MLP: 64 -> 32 (relu) -> 1, one thread per graph ----------------
__global__ void mlp_kernel(const float* __restrict__ pooled, const float* __restrict__ cnt,
                           const float* __restrict__ Wf1, const float* __restrict__ bf1,
                           const float* __restrict__ Wf2, const float* __restrict__ bf2,
                           float* __restrict__ out) {
  int g = threadIdx.x;
  if (g >= NG) return;
  float inv = 1.0f / fmaxf(cnt[g], 1.0f);
  float p[HID];
#pragma unroll
  for (int k = 0; k < HID; ++k) p[k] = pooled[(size_t)g * HID + k] * inv;
  float o = bf2[0];
  for (int j = 0; j < 32; ++j) {
    float z = bf1[j];
#pragma unroll
    for (int k = 0; k < HID; ++k) z += p[k] * Wf1[k * 32 + j];
    o += fmaxf(z, 0.0f) * Wf2[j];
  }
  out[g] = o;
}

// ---------------- host orchestration ----------------
extern "C" void kernel_launch(void* const* d_in, const int* in_sizes, int n_in,
                              void* d_out, int out_size, void* d_ws, size_t ws_size,
                              hipStream_t stream) {
  const float* x    = (const float*)d_in[0];
  const int*   ei   = (const int*)d_in[1];      // [2, NE]
  const int*   src  = ei;
  const int*   dst  = ei + NE;
  // d_in[2] edge_attr unused
  const int*   batch = (const int*)d_in[3];
  const float* W1 = (const float*)d_in[4];  const float* b1 = (const float*)d_in[5];
  const float* W2 = (const float*)d_in[6];  const float* b2 = (const float*)d_in[7];
  const float* W3 = (const float*)d_in[8];  const float* b3 = (const float*)d_in[9];
  const float* Wf1 = (const float*)d_in[10]; const float* bf1 = (const float*)d_in[11];
  const float* Wf2 = (const float*)d_in[12]; const float* bf2 = (const float*)d_in[13];
  float* out = (float*)d_out;

  const int nTot = NE + NN;                       // edges + self loops
  // workspace carve-up (256B aligned)
  char* ws = (char*)d_ws;
  size_t off = 0;
  float* dinv = (float*)(ws + off); off += ((size_t)NN * 4 + 255) & ~(size_t)255;
  float* norm = (float*)(ws + off); off += ((size_t)nTot * 4 + 255) & ~(size_t)255;
  float* buf0 = (float*)(ws + off); off += (size_t)NN * HID * 4;    // multiple of 256
  float* buf1 = (float*)(ws + off); off += (size_t)NN * HID * 4;
  float* pooled = (float*)(ws + off); off += ((size_t)NG * HID * 4 + 255) & ~(size_t)255;
  float* cnt = (float*)(ws + off); off += 256;
  (void)ws_size; (void)in_sizes; (void)n_in; (void)out_size;

  const int TB = 256;
  const long long nh = (long long)NN * HID;

  // normalization: deg -> dinv -> norm
  zero_kernel<<<cdiv(NN, TB), TB, 0, stream>>>(dinv, NN);
  degree_kernel<<<cdiv(nTot, TB), TB, 0, stream>>>(dst, dinv, NE, nTot);
  rsqrt_kernel<<<cdiv(NN, TB), TB, 0, stream>>>(dinv, NN);
  norm_kernel<<<cdiv(nTot, TB), TB, 0, stream>>>(src, dst, dinv, norm, NE, nTot);

  const unsigned gemmGrid = cdiv(NN / 16, 4);     // 4 waves per block, 1 M-tile per wave
  const unsigned scatGrid = cdiv((long long)nTot * 16, TB);

  // layer 1: x[.,16] -> buf0 ; scatter -> buf1 ; relu -> buf0
  gemm_wmma_kernel<<<gemmGrid, 128, 0, stream>>>(x, W1, buf0, NN, FIN);
  zero_kernel<<<cdiv(nh, TB), TB, 0, stream>>>(buf1, nh);
  edge_scatter_kernel<<<scatGrid, TB, 0, stream>>>(buf0, src, dst, norm, buf1, NE, nTot);
  bias_relu_kernel<<<cdiv(nh, TB), TB, 0, stream>>>(buf1, b1, buf0, nh);

  // layer 2: buf0 -> buf1 ; scatter -> buf0 ; relu -> buf1
  gemm_wmma_kernel<<<gemmGrid, 128, 0, stream>>>(buf0, W2, buf1, NN, HID);
  zero_kernel<<<cdiv(nh, TB), TB, 0, stream>>>(buf0, nh);
  edge_scatter_kernel<<<scatGrid, TB, 0, stream>>>(buf1, src, dst, norm, buf0, NE, nTot);
  bias_relu_kernel<<<cdiv(nh, TB), TB, 0, stream>>>(buf0, b2, buf1, nh);

  // layer 3: buf1 -> buf0 ; scatter -> buf1 ; relu -> buf0
  gemm_wmma_kernel<<<gemmGrid, 128, 0, stream>>>(buf1, W3, buf0, NN, HID);
  zero_kernel<<<cdiv(nh, TB), TB, 0, stream>>>(buf1, nh);
  edge_scatter_kernel<<<scatGrid, TB, 0, stream>>>(buf0, src, dst, norm, buf1, NE, nTot);
  bias_relu_kernel<<<cdiv(nh, TB), TB, 0, stream>>>(buf1, b3, buf0, nh);

  // pooling + head
  zero_kernel<<<cdiv(NG * HID + NG, TB), TB, 0, stream>>>(pooled, NG * HID);
  zero_kernel<<<1, TB, 0, stream>>>(cnt, NG);
  pool_kernel<<<cdiv(nh, TB), TB, 0, stream>>>(buf0, batch, pooled, cnt, NN);
  mlp_kernel<<<1, NG, 0, stream>>>(pooled, cnt, Wf1, bf1, Wf2, bf2, out);
}